// SSIMLoss_15015205667523
// MI455X (gfx1250) — compile-verified
//
#include <hip/hip_runtime.h>
#include <math.h>

typedef __attribute__((ext_vector_type(2))) float v2f;
typedef __attribute__((ext_vector_type(8))) float v8f;

namespace {
constexpr int H = 384, W = 512;
constexpr int HV = H - 10, WV = W - 10;          // 374 x 502 valid outputs
constexpr int PLANES = 16 * 3;                   // N*C independent planes
constexpr int TILE_H = 32;                       // output tile: 32 rows x 16 cols
constexpr int TILES_X = (WV + 15) / 16;          // 32
constexpr int TILES_Y = (HV + TILE_H - 1) / TILE_H;  // 12
constexpr int TILES_PER_PLANE = TILES_X * TILES_Y;   // 384
constexpr int TOTAL_TILES = TILES_PER_PLANE * PLANES;  // 18432
constexpr int WPB = 8;                           // waves (tiles) per block
constexpr int BLOCK = WPB * 32;
constexpr int NBLOCKS = TOTAL_TILES / WPB;       // 2304
constexpr float C1 = 0.0001f;                    // (K1*range)^2
constexpr float C2 = 0.0009f;                    // (K2*range)^2
}

static __device__ inline v8f wmma4(v2f a, v2f b, v8f c) {
    // D = A(16x4,f32) x B(4x16,f32) + C(16x16,f32)
    return __builtin_amdgcn_wmma_f32_16x16x4_f32(
        false, a, false, b, (short)0, c, false, false);
}

__global__ __launch_bounds__(BLOCK) void ssim_tile_kernel(
    const float* __restrict__ pred, const float* __restrict__ targ,
    float* __restrict__ partial)
{
    __shared__ float wpad[64];             // zero-padded gaussian: wpad[16+i] = w[i]
    __shared__ float hbuf[WPB][24 * 32];   // per-wave scratch, row-pair packed:
                                           // h[m][n] at [(m>>1)*32 + n*2 + (m&1)]
    __shared__ float red[BLOCK];

    const int lane = threadIdx.x & 31;
    const int wv   = threadIdx.x >> 5;
    const int hi   = lane >> 4;            // lane half (0/1)
    const int ln   = lane & 15;

    // ---- Build padded Gaussian LUT once per block ----
    {
        int t = threadIdx.x;
        if (t < 64) {
            float v = 0.f;
            if (t >= 16 && t <= 26) {
                float d = (float)(t - 21);
                float g = expf(-(d * d) / 4.5f);       // runtime, once per block
                float s = 0.f;
                #pragma unroll
                for (int i = 0; i < 11; i++) {         // constant-folds to a literal
                    float di = (float)(i - 5);
                    s += expf(-(di * di) / 4.5f);
                }
                v = g / s;
            }
            wpad[t] = v;
        }
    }
    __syncthreads();

    const int tile  = blockIdx.x * WPB + wv;
    const int plane = tile / TILES_PER_PLANE;
    const int tr    = tile % TILES_PER_PLANE;
    const int y0    = (tr / TILES_X) * TILE_H;
    const int x0    = (tr % TILES_X) * 16;
    const unsigned pbase = (unsigned)plane * (H * W);

    // Per-lane constant operand table via LDS lookup.
    // For K-chunk c, vgpr j: K = 4c+j+2*hi. Used both as B (Bh[k][n]=w[k-n],
    // n=ln) and as A (Wv[m][k]=w[k-m], m=ln): both are w[K - ln].
    const int tbase = 16 + 2 * hi - ln;    // in [1, 18]
    v2f T[7];
    #pragma unroll
    for (int c = 0; c < 7; c++) {
        T[c].x = wpad[tbase + 4 * c];
        T[c].y = wpad[tbase + 4 * c + 1];
    }

    // ---- Stage 1: horizontal blur of 5 maps (input rows 0..47 of tile) ----
    v8f accH[3][5];
    #pragma unroll
    for (int mt = 0; mt < 3; mt++)
        #pragma unroll
        for (int m = 0; m < 5; m++)
            accH[mt][m] = (v8f){0.f,0.f,0.f,0.f,0.f,0.f,0.f,0.f};

    // doClamp is a literal at each call site -> compiler clones; fast path
    // (interior tiles, wave-uniform branch) folds offsets into load immediates.
    auto stage1 = [&](bool doClamp) {
        #pragma unroll
        for (int mt = 0; mt < 3; mt++) {
            int yy = y0 + ln + 16 * mt;
            if (doClamp && yy > H - 1) yy = H - 1;   // garbage h-rows >41 unused
            const unsigned rowb = pbase + (unsigned)yy * W + (unsigned)(x0 + 2 * hi);
            #pragma unroll
            for (int c = 0; c < 7; c++) {
                unsigned off = rowb + 4u * c;
                if (doClamp) {
                    int xx = x0 + 4 * c + 2 * hi;    // clamped cols: zero wt / masked
                    if (xx > W - 2) xx = W - 2;
                    off = pbase + (unsigned)yy * W + (unsigned)xx;
                }
                v2f X = *(const v2f*)(pred + off);
                v2f Y = *(const v2f*)(targ + off);
                v2f XX = X * X, YY = Y * Y, XY = X * Y;
                accH[mt][0] = wmma4(X,  T[c], accH[mt][0]);
                accH[mt][1] = wmma4(Y,  T[c], accH[mt][1]);
                accH[mt][2] = wmma4(XX, T[c], accH[mt][2]);
                accH[mt][3] = wmma4(YY, T[c], accH[mt][3]);
                accH[mt][4] = wmma4(XY, T[c], accH[mt][4]);
            }
        }
    };
    const bool interior = (x0 + 26 <= W - 2) && (y0 + 47 <= H - 1);
    if (interior) stage1(false); else stage1(true);

    // ---- Stage 2: vertical blur (per map: D-layout -> LDS -> B-layout) ----
    // Row-pair packing makes every transfer a conflict-friendly b64 op.
    float* hb = hbuf[wv];
    v8f accV[2][5];
    #pragma unroll
    for (int m = 0; m < 5; m++) {
        #pragma unroll
        for (int mt = 0; mt < 3; mt++)
            #pragma unroll
            for (int r = 0; r < 8; r += 2) {
                // rows mrow=r+8hi+16mt (even) and mrow+1
                int prow = (r >> 1) + 4 * hi + 8 * mt;   // mrow>>1
                v2f p; p.x = accH[mt][m][r]; p.y = accH[mt][m][r + 1];
                *(v2f*)&hb[prow * 32 + ln * 2] = p;      // ds_store_b64
            }
        // same-wave LDS RAW: compiler inserts s_wait_dscnt; no barrier needed
        #pragma unroll
        for (int o = 0; o < 2; o++) {               // two 16-row output sub-tiles
            v8f acc = (v8f){0.f,0.f,0.f,0.f,0.f,0.f,0.f,0.f};
            #pragma unroll
            for (int c = 0; c < 7; c++) {
                // B rows k0=16o+4c+2hi (even), k0+1 -> one ds_load_b64
                v2f b = *(const v2f*)&hb[(8 * o + 2 * c + hi) * 32 + ln * 2];
                acc = wmma4(T[c], b, acc);
            }
            accV[o][m] = acc;
        }
    }

    // ---- Pointwise SSIM + masked accumulation (single fast divide) ----
    float lsum = 0.f;
    #pragma unroll
    for (int o = 0; o < 2; o++) {
        #pragma unroll
        for (int r = 0; r < 8; r++) {
            int m = 16 * o + r + 8 * hi;         // output row within tile (D layout)
            int yo = y0 + m, xo = x0 + ln;
            float mu1 = accV[o][0][r], mu2 = accV[o][1][r];
            float sxx = accV[o][2][r], syy = accV[o][3][r], sxy = accV[o][4][r];
            float mu1s = mu1 * mu1, mu2s = mu2 * mu2, mu12 = mu1 * mu2;
            float s1 = sxx - mu1s, s2 = syy - mu2s, s12 = sxy - mu12;
            float num = (2.f * mu12 + C1) * (2.f * s12 + C2);
            float den = (mu1s + mu2s + C1) * (s1 + s2 + C2);
            float ssim = __fdividef(num, den);
            if (yo < HV && xo < WV) lsum += ssim;
        }
    }

    // ---- Deterministic block reduction ----
    red[threadIdx.x] = lsum;
    __syncthreads();
    for (int off = BLOCK / 2; off > 0; off >>= 1) {
        if (threadIdx.x < off) red[threadIdx.x] += red[threadIdx.x + off];
        __syncthreads();
    }
    if (threadIdx.x == 0) partial[blockIdx.x] = red[0];
}

__global__ __launch_bounds__(256) void ssim_finalize_kernel(
    const float* __restrict__ partial, float* __restrict__ out)
{
    __shared__ float red[256];
    float s = 0.f;
    for (int i = threadIdx.x; i < NBLOCKS; i += 256) s += partial[i];
    red[threadIdx.x] = s;
    __syncthreads();
    for (int off = 128; off > 0; off >>= 1) {
        if (threadIdx.x < off) red[threadIdx.x] += red[threadIdx.x + off];
        __syncthreads();
    }
    if (threadIdx.x == 0)
        out[0] = 1.0f - red[0] / (float)(PLANES * HV * WV);
}

extern "C" void kernel_launch(void* const* d_in, const int* in_sizes, int n_in,
                              void* d_out, int out_size, void* d_ws, size_t ws_size,
                              hipStream_t stream) {
    (void)in_sizes; (void)n_in; (void)out_size; (void)ws_size;
    const float* pred = (const float*)d_in[0];
    const float* targ = (const float*)d_in[1];
    float* partial = (float*)d_ws;            // NBLOCKS floats, overwritten every call
    float* out = (float*)d_out;

    ssim_tile_kernel<<<NBLOCKS, BLOCK, 0, stream>>>(pred, targ, partial);
    ssim_finalize_kernel<<<1, 256, 0, stream>>>(partial, out);
}